// SpikeFP32Embedding_76312978916091
// MI455X (gfx1250) — compile-verified
//
#include <hip/hip_runtime.h>

// ---------------------------------------------------------------------------
// SpikeFP32Embedding gather: out[i, :, :] = weight_pulse[token_ids[i], :, :]
// Row = 128*32 f32 = 4096 floats = 16 KB. Pure HBM-bandwidth copy problem
// (512 MB total traffic -> ~22us at 23.3 TB/s). Strategy: wave-per-row,
// async global->LDS->global staging via the CDNA5 ASYNCcnt engine.
// Reads stay temporal (token repeats hit the 192MB L2); streamed output
// stores are non-temporal so they don't evict the gather working set.
// ---------------------------------------------------------------------------

#define AS1 __attribute__((address_space(1)))
#define AS3 __attribute__((address_space(3)))

typedef int   v4i __attribute__((ext_vector_type(4)));
typedef float v4f __attribute__((ext_vector_type(4)));

#if defined(__AMDGCN__) &&                                            \
    __has_builtin(__builtin_amdgcn_global_load_async_to_lds_b128) &&  \
    __has_builtin(__builtin_amdgcn_global_store_async_from_lds_b128)
#define HAVE_ASYNC_LDS 1
#else
#define HAVE_ASYNC_LDS 0
#endif

// gfx12+ cpol encoding: TH[2:0], SCOPE[4:3]. Store TH_NT = 1 (non-temporal).
#define CPOL_RT     0
#define CPOL_ST_NT  1

__device__ __forceinline__ void wait_async_zero() {
#if defined(__AMDGCN__)
#if __has_builtin(__builtin_amdgcn_s_wait_asynccnt)
  __builtin_amdgcn_s_wait_asynccnt(0);
#else
  asm volatile("s_wait_asynccnt 0" ::: "memory");
#endif
#endif
}

constexpr int ROW_FLOATS      = 128 * 32;              // 4096 floats = 16 KB
constexpr int WAVES_PER_BLOCK = 8;                     // 256 threads, wave32
constexpr int LANES           = 32;
constexpr int CHUNK_FLOATS    = LANES * 4;             // 128 floats / async op (512 B)
constexpr int STAGE_FLOATS    = 1024;                  // 4 KB LDS slice per wave
constexpr int CHUNKS_PER_STAGE = STAGE_FLOATS / CHUNK_FLOATS;  // 8
constexpr int STAGES          = ROW_FLOATS / STAGE_FLOATS;     // 4

__global__ __launch_bounds__(WAVES_PER_BLOCK * LANES)
void SpikeFP32Embedding_gather_kernel(const int* __restrict__ token_ids,
                                      const float* __restrict__ table,
                                      float* __restrict__ out,
                                      int n_tokens) {
  const int wave = threadIdx.x >> 5;
  const int lane = threadIdx.x & (LANES - 1);
  const int tokenIdx = blockIdx.x * WAVES_PER_BLOCK + wave;
  if (tokenIdx >= n_tokens) return;   // whole-wave early exit; EXEC stays all-1s

  const int tok = token_ids[tokenIdx];
  const float* __restrict__ src = table + (size_t)tok      * ROW_FLOATS;
  float*       __restrict__ dst = out   + (size_t)tokenIdx * ROW_FLOATS;

#if HAVE_ASYNC_LDS
  __shared__ float lds[WAVES_PER_BLOCK][STAGE_FLOATS];   // 32 KB static LDS
  for (int s = 0; s < STAGES; ++s) {
    const float* sstage = src + s * STAGE_FLOATS;
    float*       dstage = dst + s * STAGE_FLOATS;
    // 8 outstanding b128 async loads: 4 KB in flight per wave, no VGPR traffic.
#pragma unroll
    for (int c = 0; c < CHUNKS_PER_STAGE; ++c) {
      __builtin_amdgcn_global_load_async_to_lds_b128(
          (AS1 v4i*)(sstage + c * CHUNK_FLOATS + lane * 4),
          (AS3 v4i*)(&lds[wave][c * CHUNK_FLOATS + lane * 4]),
          /*offset=*/0, /*cpol=*/CPOL_RT);
    }
    wait_async_zero();   // loads landed in LDS
#pragma unroll
    for (int c = 0; c < CHUNKS_PER_STAGE; ++c) {
      __builtin_amdgcn_global_store_async_from_lds_b128(
          (AS1 v4i*)(dstage + c * CHUNK_FLOATS + lane * 4),
          (AS3 v4i*)(&lds[wave][c * CHUNK_FLOATS + lane * 4]),
          /*offset=*/0, /*cpol=*/CPOL_ST_NT);
    }
    wait_async_zero();   // stores drained before LDS slice is reused
  }
#else
  // Fallback: direct b128 copy through VGPRs; nontemporal stores keep the
  // streamed output from evicting the gather rows cached in L2.
  const v4f* __restrict__ s4 = (const v4f*)src;
  v4f*       __restrict__ d4 = (v4f*)dst;
#pragma unroll 4
  for (int i = lane; i < ROW_FLOATS / 4; i += LANES) {
    v4f v = s4[i];
    __builtin_nontemporal_store(v, d4 + i);
  }
#endif
}

extern "C" void kernel_launch(void* const* d_in, const int* in_sizes, int n_in,
                              void* d_out, int out_size, void* d_ws, size_t ws_size,
                              hipStream_t stream) {
  const int*   token_ids = (const int*)d_in[0];    // [8*2048]
  const float* table     = (const float*)d_in[1];  // [32768, 128, 32]
  float*       out       = (float*)d_out;          // [8*2048, 128, 32]

  const int n_tokens = in_sizes[0];                // 16384
  const int blocks = (n_tokens + WAVES_PER_BLOCK - 1) / WAVES_PER_BLOCK;

  SpikeFP32Embedding_gather_kernel<<<blocks, WAVES_PER_BLOCK * LANES, 0, stream>>>(
      token_ids, table, out, n_tokens);
}